// Transformer_66786741453066
// MI455X (gfx1250) — compile-verified
//
#include <hip/hip_runtime.h>
#include <hip/hip_bf16.h>

typedef __attribute__((ext_vector_type(2))) float v2f;
typedef __attribute__((ext_vector_type(8))) float v8f;

#define OUT_H 192
#define OUT_W 192
#define IMG_H 256
#define IMG_W 256
#define NCH   3
#define NPIX  (OUT_H * OUT_W)          // 36864 pixels per batch
#define NTILES (NPIX / 16)             // 2304 16-pixel tiles
#define WAVES_PER_BLOCK 4
#define TILES_PER_WAVE  8
#define TILES_PER_BLOCK (WAVES_PER_BLOCK * TILES_PER_WAVE)  // 32
#define BLOCKS_X (NTILES / TILES_PER_BLOCK)                 // 72
#define BATCH_PAIRS 64                                       // B=128 / 2

// One wave: WMMA f32 16x16x4 computes affine source coords for 16 pixels x 2
// batches per issue (A = two thetas in rows {0,1,8,9}, B = grid columns).
// Each lane then does the bilinear 4-tap gather for its (pixel, batch).
__global__ __launch_bounds__(128)
void stn_wmma_kernel(const float* __restrict__ fmap,
                     const float* __restrict__ theta,
                     float* __restrict__ out) {
    const int tid  = threadIdx.x;
    const int wave = tid >> 5;
    const int lane = tid & 31;
    const int half = lane >> 4;    // 0: batch b0 / K-lo, 1: batch b1 / K-hi
    const int r    = lane & 15;    // A-row (lanes 0-15) / B,D column N
    const int bp   = blockIdx.y;   // batch pair index
    const int b    = bp * 2 + half;

    // ---- Build A matrix fragment (constant for the whole wave loop) ----
    // A rows: 0 -> theta[b0] row0, 1 -> theta[b0] row1,
    //         8 -> theta[b1] row0, 9 -> theta[b1] row1, rest zero.
    // Lanes 0-15 hold K=0,1 ; lanes 16-31 hold K=2,3 (K=3 column is zero).
    const float* th = theta + bp * 12;
    float ax = 0.0f, ay = 0.0f;
    int sel = -1;
    if      (r == 0) sel = 0;   // b0 row0
    else if (r == 1) sel = 1;   // b0 row1
    else if (r == 8) sel = 2;   // b1 row0
    else if (r == 9) sel = 3;   // b1 row1
    if (sel >= 0) {
        const float* tr = th + (sel >> 1) * 6 + (sel & 1) * 3;
        if (half == 0) { ax = tr[0]; ay = tr[1]; }   // K=0,1: scale terms
        else           { ax = tr[2]; ay = 0.0f;  }   // K=2: translation; K=3: 0
    }
    v2f A; A.x = ax; A.y = ay;

    const float inv191 = 2.0f / 191.0f;  // linspace(-1,1,192) step
    const float* __restrict__ fb = fmap + (size_t)b * (IMG_H * IMG_W * NCH);
    float*       __restrict__ ob = out  + (size_t)b * (NPIX * NCH);

    const int tile0 = blockIdx.x * TILES_PER_BLOCK + wave * TILES_PER_WAVE;

    #pragma unroll 2
    for (int i = 0; i < TILES_PER_WAVE; ++i) {
        const int tile = tile0 + i;
        const int p  = tile * 16 + r;      // this lane's output pixel (column N)
        const int oy = p / OUT_W;
        const int ox = p - oy * OUT_W;

        // ---- Build B matrix fragment: column N = (x_t, y_t, 1, 0)^T ----
        // Lanes 0-15: rows K=0,1 (grid coords); lanes 16-31: rows K=2,3 (1, 0).
        v2f Bv;
        if (half == 0) {
            Bv.x = fmaf((float)ox, inv191, -1.0f);
            Bv.y = fmaf((float)oy, inv191, -1.0f);
        } else {
            Bv.x = 1.0f;
            Bv.y = 0.0f;
        }

        // ---- D = A x B : v_wmma_f32_16x16x4_f32 ----
        v8f Cacc = {};
        Cacc = __builtin_amdgcn_wmma_f32_16x16x4_f32(
            /*neg_a=*/false, A, /*neg_b=*/false, Bv,
            /*c_mod=*/(short)0, Cacc, /*reuse_a=*/false, /*reuse_b=*/false);

        // D layout: lane (half,N) VGPR0 = D[8*half+0][N] = x_s,
        //                        VGPR1 = D[8*half+1][N] = y_s  for batch b.
        const float xs = Cacc[0];
        const float ys = Cacc[1];

        // ---- Bilinear sample, exactly as the reference ----
        const float x = 0.5f * (xs + 1.0f) * 254.0f;   // (W-1)-1 = 254
        const float y = 0.5f * (ys + 1.0f) * 254.0f;   // (H-1)-1 = 254
        const float xf = floorf(x);
        const float yf = floorf(y);
        const int x0 = (int)xf;
        const int y0 = (int)yf;
        const int x0c = min(max(x0,     0), IMG_W - 1);
        const int x1c = min(max(x0 + 1, 0), IMG_W - 1);
        const int y0c = min(max(y0,     0), IMG_H - 1);
        const int y1c = min(max(y0 + 1, 0), IMG_H - 1);

        const float x0f = (float)x0c, x1f = (float)x1c;
        const float y0f = (float)y0c, y1f = (float)y1c;
        const float wa = (x1f - x) * (y1f - y);
        const float wb = (x1f - x) * (y - y0f);
        const float wc = (x - x0f) * (y1f - y);
        const float wd = (x - x0f) * (y - y0f);

        const float* __restrict__ pa = fb + (y0c * IMG_W + x0c) * NCH;
        const float* __restrict__ pb = fb + (y1c * IMG_W + x0c) * NCH;
        const float* __restrict__ pc = fb + (y0c * IMG_W + x1c) * NCH;
        const float* __restrict__ pd = fb + (y1c * IMG_W + x1c) * NCH;
        float* __restrict__ po = ob + p * NCH;

        #pragma unroll
        for (int ch = 0; ch < NCH; ++ch) {
            po[ch] = wa * pa[ch] + wb * pb[ch] + wc * pc[ch] + wd * pd[ch];
        }
    }
}

extern "C" void kernel_launch(void* const* d_in, const int* in_sizes, int n_in,
                              void* d_out, int out_size, void* d_ws, size_t ws_size,
                              hipStream_t stream) {
    (void)in_sizes; (void)n_in; (void)out_size; (void)d_ws; (void)ws_size;
    const float* fmap  = (const float*)d_in[0];   // (128,256,256,3) f32
    const float* theta = (const float*)d_in[1];   // (128,6) f32
    float* out = (float*)d_out;                   // (128,192,192,3) f32

    dim3 grid(BLOCKS_X, BATCH_PAIRS);   // 72 x 64
    dim3 block(WAVES_PER_BLOCK * 32);   // 128 threads = 4 wave32
    stn_wmma_kernel<<<grid, block, 0, stream>>>(fmap, theta, out);
}